// GCN_PointNet_41901700940152
// MI455X (gfx1250) — compile-verified
//
#include <hip/hip_runtime.h>
#include <hip/hip_bf16.h>
#include <math.h>

typedef __attribute__((ext_vector_type(2))) float v2f;
typedef __attribute__((ext_vector_type(8))) float v8f;

#define EPS 1e-5f

// ---------------------------------------------------------------------------
// WMMA GEMM: H(N x C) = X(N x K) @ W^T + bias, W is (C x K) row-major.
// One wave computes one 16x16 output tile with V_WMMA_F32_16X16X4_F32.
// K is compile-time -> fully unrolled WMMA chain, branch-free K loop.
// A rows / B cols are CLAMPED (not predicated): out-of-range lanes only
// pollute D rows/cols that the store path discards. For K%4 != 0 the padded
// k slot is loaded clamped-in-bounds and zeroed with a value select
// (v_cndmask), never a predicated load. Stores take a wave-uniform fast path
// for interior tiles (no per-lane exec manipulation).
// ---------------------------------------------------------------------------
template <int K>
__global__ __launch_bounds__(32) void gemm_bias_wmma(
        const float* __restrict__ X,
        const float* __restrict__ W,
        const float* __restrict__ bias,
        float* __restrict__ H,
        int N, int C) {
    const int lane  = threadIdx.x;      // 0..31, one wave per block
    const int l16   = lane & 15;
    const int khalf = lane >> 4;        // 0 or 1
    const int m0 = blockIdx.x * 16;
    const int n0 = blockIdx.y * 16;

    const int gm = min(m0 + l16, N - 1);    // clamped A row
    const int gn = min(n0 + l16, C - 1);    // clamped B col
    const float* __restrict__ Xrow = X + (size_t)gm * K;
    const float* __restrict__ Wrow = W + (size_t)gn * K;

    v8f acc = {};
    constexpr int KP = (K + 3) & ~3;        // pad K to multiple of 4
#pragma unroll
    for (int kb = 0; kb < KP; kb += 4) {
        const int k0 = kb + 2 * khalf;
        v2f a, b;
        if constexpr ((K & 3) == 0) {
            a = *(const v2f*)(Xrow + k0);    // global_load_b64
            b = *(const v2f*)(Wrow + k0);
        } else {
            // clamp index -> unconditional load; zero padded slot via select
            const int ka = min(k0, K - 1);
            const int kc = min(k0 + 1, K - 1);
            const float ax = Xrow[ka], ay = Xrow[kc];
            const float bx = Wrow[ka], by = Wrow[kc];
            a.x = (k0     < K) ? ax : 0.0f;
            a.y = (k0 + 1 < K) ? ay : 0.0f;
            b.x = (k0     < K) ? bx : 0.0f;
            b.y = (k0 + 1 < K) ? by : 0.0f;
        }
        acc = __builtin_amdgcn_wmma_f32_16x16x4_f32(
            /*neg_a=*/false, a, /*neg_b=*/false, b,
            /*c_mod=*/(short)0, acc, /*reuse_a=*/false, /*reuse_b=*/false);
    }

    const int col     = n0 + l16;
    const int rowbase = m0 + khalf * 8;
    if (m0 + 16 <= N && n0 + 16 <= C) {
        // interior tile: wave-uniform, all stores unguarded
        const float bv = bias[col];
        float* __restrict__ p = H + (size_t)rowbase * C + col;
#pragma unroll
        for (int r = 0; r < 8; ++r) p[(size_t)r * C] = acc[r] + bv;
    } else if (col < C) {
        const float bv = bias[col];
#pragma unroll
        for (int r = 0; r < 8; ++r) {
            const int row = rowbase + r;
            if (row < N) H[row * C + col] = acc[r] + bv;
        }
    }
}

// ---------------------------------------------------------------------------
// BatchNorm1d (training stats over N nodes) + ReLU, in place. One block/channel.
// ---------------------------------------------------------------------------
__global__ void bn_relu(float* __restrict__ H,
                        const float* __restrict__ g,
                        const float* __restrict__ be,
                        int N, int C) {
    const int c = blockIdx.x;
    __shared__ float s1[256];
    __shared__ float s2[256];
    float sum = 0.0f, sq = 0.0f;
    for (int r = threadIdx.x; r < N; r += blockDim.x) {
        const float v = H[r * C + c];
        sum += v; sq += v * v;
    }
    s1[threadIdx.x] = sum; s2[threadIdx.x] = sq;
    __syncthreads();
    for (int off = 128; off > 0; off >>= 1) {
        if ((int)threadIdx.x < off) {
            s1[threadIdx.x] += s1[threadIdx.x + off];
            s2[threadIdx.x] += s2[threadIdx.x + off];
        }
        __syncthreads();
    }
    const float inv_n = 1.0f / (float)N;
    const float mu    = s1[0] * inv_n;
    const float var   = s2[0] * inv_n - mu * mu;   // biased, like torch BN
    const float scale = g[c] * rsqrtf(var + EPS);
    const float shift = be[c];
    for (int r = threadIdx.x; r < N; r += blockDim.x) {
        const float v = (H[r * C + c] - mu) * scale + shift;
        H[r * C + c] = fmaxf(v, 0.0f);
    }
}

// ---------------------------------------------------------------------------
// Zero-fill
// ---------------------------------------------------------------------------
__global__ void zero_f32(float* __restrict__ p, int n) {
    const int i = blockIdx.x * blockDim.x + threadIdx.x;
    if (i < n) p[i] = 0.0f;
}

// ---------------------------------------------------------------------------
// Weighted segment-sum over edges: agg[dst] += ew * X[src].
// One thread per (edge, channel): coalesced reads, one f32 atomic each.
// ---------------------------------------------------------------------------
__global__ void seg_agg(const float* __restrict__ X,
                        const int* __restrict__ src,
                        const int* __restrict__ dst,
                        const float* __restrict__ ew,
                        float* __restrict__ agg,
                        int E, int C) {
    const int i = blockIdx.x * blockDim.x + threadIdx.x;
    if (i >= E * C) return;
    const int e = i / C;
    const int c = i - e * C;
    const float w = ew[e];
    if (w == 0.0f) return;                  // masked-out edge
    atomicAdd(&agg[dst[e] * C + c], w * X[src[e] * C + c]);
}

// ---------------------------------------------------------------------------
// GraphConv scorer: score[n] = agg[n].wrel + x[n].wroot + b
// ---------------------------------------------------------------------------
__global__ void score_nodes(const float* __restrict__ agg,
                            const float* __restrict__ X,
                            const float* __restrict__ wrel,
                            const float* __restrict__ wroot,
                            const float* __restrict__ bptr,
                            float* __restrict__ score,
                            int N, int C) {
    const int n = blockIdx.x * blockDim.x + threadIdx.x;
    if (n >= N) return;
    float s = bptr[0];
    for (int c = 0; c < C; ++c)
        s += agg[n * C + c] * wrel[c] + X[n * C + c] * wroot[c];
    score[n] = s;
}

// ---------------------------------------------------------------------------
// Single-block bitonic top-k (descending, tie-break lower index first, like
// lax.top_k) + tanh-scaled gather + new-index table (-1 for dropped nodes).
// blockDim.x must be a power of two >= N (1024 for pool1, 512 for pool2).
// ---------------------------------------------------------------------------
__global__ void topk_pool(const float* __restrict__ score,
                          const float* __restrict__ Xin,
                          float* __restrict__ Xout,
                          int* __restrict__ ni,
                          int N, int k, int C) {
    __shared__ float ss[1024];
    __shared__ int   si[1024];
    const int t = threadIdx.x;
    const int M = blockDim.x;

    ss[t] = (t < N) ? score[t] : -INFINITY;
    si[t] = t;
    if (t < N) ni[t] = -1;
    __syncthreads();

    for (int kk = 2; kk <= M; kk <<= 1) {
        for (int j = kk >> 1; j > 0; j >>= 1) {
            const int ixj = t ^ j;
            if (ixj > t) {
                const float sa = ss[t],  sb = ss[ixj];
                const int   ia = si[t],  ib = si[ixj];
                // "a ranks earlier" in descending order with stable ties
                const bool aFirst = (sa > sb) || (sa == sb && ia < ib);
                const bool descSeg = ((t & kk) == 0);
                if (descSeg ? !aFirst : aFirst) {
                    ss[t] = sb; ss[ixj] = sa;
                    si[t] = ib; si[ixj] = ia;
                }
            }
            __syncthreads();
        }
    }

    if (t < k) {
        const int p = si[t];
        ni[p] = t;
        const float scale = tanhf(ss[t]);
        for (int c = 0; c < C; ++c)
            Xout[t * C + c] = Xin[p * C + c] * scale;
    }
}

// ---------------------------------------------------------------------------
// Re-index edges after pooling; invalid edges get weight 0 and indices 0.
// ---------------------------------------------------------------------------
__global__ void remap_edges(const int* __restrict__ srcIn,
                            const int* __restrict__ dstIn,
                            const float* __restrict__ ewIn,
                            const int* __restrict__ ni,
                            int* __restrict__ srcOut,
                            int* __restrict__ dstOut,
                            float* __restrict__ ewOut,
                            int E) {
    const int e = blockIdx.x * blockDim.x + threadIdx.x;
    if (e >= E) return;
    const int ns = ni[srcIn[e]];
    const int nd = ni[dstIn[e]];
    const float w = ewIn[e];
    const bool valid = (ns >= 0) && (nd >= 0) && (w != 0.0f);
    srcOut[e] = valid ? ns : 0;
    dstOut[e] = valid ? nd : 0;
    ewOut[e]  = valid ? w : 0.0f;
}

// ---------------------------------------------------------------------------
// Head: global max pool (100 x 200) -> Linear(200,128)+ReLU ->
// Linear(128,2)+ReLU -> log_softmax. One 256-thread block.
// ---------------------------------------------------------------------------
__global__ void head(const float* __restrict__ X,      // (100, 200)
                     const float* __restrict__ w3a,    // (128, 200)
                     const float* __restrict__ b3a,
                     const float* __restrict__ w3b,    // (2, 128)
                     const float* __restrict__ b3b,
                     float* __restrict__ out) {
    __shared__ float gbuf[200];
    __shared__ float h3[128];
    __shared__ float ov[2];
    const int t = threadIdx.x;

    if (t < 200) {
        float m = -INFINITY;
        for (int r = 0; r < 100; ++r) m = fmaxf(m, X[r * 200 + t]);
        gbuf[t] = m;
    }
    __syncthreads();
    if (t < 128) {
        float s = b3a[t];
        for (int c = 0; c < 200; ++c) s += gbuf[c] * w3a[t * 200 + c];
        h3[t] = fmaxf(s, 0.0f);
    }
    __syncthreads();
    if (t < 2) {
        float s = b3b[t];
        for (int c = 0; c < 128; ++c) s += h3[c] * w3b[t * 128 + c];
        ov[t] = fmaxf(s, 0.0f);
    }
    __syncthreads();
    if (t == 0) {
        const float m   = fmaxf(ov[0], ov[1]);
        const float lse = m + logf(expf(ov[0] - m) + expf(ov[1] - m));
        out[0] = ov[0] - lse;
        out[1] = ov[1] - lse;
    }
}

// ---------------------------------------------------------------------------
// Host-side orchestration
// ---------------------------------------------------------------------------
extern "C" void kernel_launch(void* const* d_in, const int* in_sizes, int n_in,
                              void* d_out, int out_size, void* d_ws, size_t ws_size,
                              hipStream_t stream) {
    (void)in_sizes; (void)n_in; (void)out_size; (void)ws_size;
    constexpr int N0 = 1000, K1 = 500, K2 = 100, E = 32000;

    const float* x     = (const float*)d_in[0];
    const int*   ei    = (const int*)d_in[1];
    const int*   src0  = ei;
    const int*   dst0  = ei + E;
    const float* ea    = (const float*)d_in[2];
    const float* w0  = (const float*)d_in[3];
    const float* b0  = (const float*)d_in[4];
    const float* g0  = (const float*)d_in[5];
    const float* be0 = (const float*)d_in[6];
    const float* w1  = (const float*)d_in[7];
    const float* b1  = (const float*)d_in[8];
    const float* g1  = (const float*)d_in[9];
    const float* be1 = (const float*)d_in[10];
    const float* w2a  = (const float*)d_in[11];
    const float* b2a  = (const float*)d_in[12];
    const float* g2a  = (const float*)d_in[13];
    const float* be2a = (const float*)d_in[14];
    const float* w2b  = (const float*)d_in[15];
    const float* b2b  = (const float*)d_in[16];
    const float* g2b  = (const float*)d_in[17];
    const float* be2b = (const float*)d_in[18];
    const float* w3a  = (const float*)d_in[19];
    const float* b3a  = (const float*)d_in[20];
    const float* w3b  = (const float*)d_in[21];
    const float* b3b  = (const float*)d_in[22];
    const float* p1_wrel  = (const float*)d_in[23];
    const float* p1_wroot = (const float*)d_in[24];
    const float* p1_b     = (const float*)d_in[25];
    const float* p2_wrel  = (const float*)d_in[26];
    const float* p2_wroot = (const float*)d_in[27];
    const float* p2_b     = (const float*)d_in[28];

    // Workspace carve (fp32 region first, then ints)
    float* f = (float*)d_ws;
    float* H0     = f; f += N0 * 64;       // 64000
    float* H1     = f; f += N0 * 64;       // 64000
    float* agg1   = f; f += N0 * 64;       // 64000
    float* score1 = f; f += 1024;
    float* X2     = f; f += K1 * 64;       // 32000
    float* H2a    = f; f += K1 * 128;      // 64000
    float* H2b    = f; f += K1 * 200;      // 100000
    float* agg2   = f; f += K1 * 200;      // 100000
    float* score2 = f; f += 512;
    float* X3     = f; f += K2 * 200;      // 20000
    float* ew2    = f; f += E;             // 32000
    int* ip   = (int*)f;
    int* ni1  = ip; ip += 1024;
    int* ni2  = ip; ip += 512;
    int* src2 = ip; ip += E;
    int* dst2 = ip; ip += E;

    const dim3 wave(32, 1, 1);

    // ---- stage 1: input Conv(3->64)+BN+ReLU, mlp_first Conv(64->64)+BN+ReLU
    gemm_bias_wmma<3><<<dim3((N0 + 15) / 16, 64 / 16), wave, 0, stream>>>(x,  w0, b0, H0, N0, 64);
    bn_relu<<<64, 256, 0, stream>>>(H0, g0, be0, N0, 64);
    gemm_bias_wmma<64><<<dim3((N0 + 15) / 16, 64 / 16), wave, 0, stream>>>(H0, w1, b1, H1, N0, 64);
    bn_relu<<<64, 256, 0, stream>>>(H1, g1, be1, N0, 64);

    // ---- SAG pool 1: 1000 -> 500
    zero_f32<<<(N0 * 64 + 255) / 256, 256, 0, stream>>>(agg1, N0 * 64);
    seg_agg<<<(E * 64 + 255) / 256, 256, 0, stream>>>(H1, src0, dst0, ea, agg1, E, 64);
    score_nodes<<<(N0 + 255) / 256, 256, 0, stream>>>(agg1, H1, p1_wrel, p1_wroot, p1_b,
                                                      score1, N0, 64);
    topk_pool<<<1, 1024, 0, stream>>>(score1, H1, X2, ni1, N0, K1, 64);
    remap_edges<<<(E + 255) / 256, 256, 0, stream>>>(src0, dst0, ea, ni1,
                                                     src2, dst2, ew2, E);

    // ---- stage 2: Conv(64->128)+BN+ReLU, Conv(128->200)+BN+ReLU
    gemm_bias_wmma<64><<<dim3((K1 + 15) / 16, 128 / 16), wave, 0, stream>>>(X2,  w2a, b2a, H2a, K1, 128);
    bn_relu<<<128, 256, 0, stream>>>(H2a, g2a, be2a, K1, 128);
    gemm_bias_wmma<128><<<dim3((K1 + 15) / 16, (200 + 15) / 16), wave, 0, stream>>>(H2a, w2b, b2b, H2b, K1, 200);
    bn_relu<<<200, 256, 0, stream>>>(H2b, g2b, be2b, K1, 200);

    // ---- SAG pool 2: 500 -> 100
    zero_f32<<<(K1 * 200 + 255) / 256, 256, 0, stream>>>(agg2, K1 * 200);
    seg_agg<<<(E * 200 + 255) / 256, 256, 0, stream>>>(H2b, src2, dst2, ew2, agg2, E, 200);
    score_nodes<<<(K1 + 255) / 256, 256, 0, stream>>>(agg2, H2b, p2_wrel, p2_wroot, p2_b,
                                                      score2, K1, 200);
    topk_pool<<<1, 512, 0, stream>>>(score2, H2b, X3, ni2, K1, K2, 200);

    // ---- head: max pool + MLP + log_softmax
    head<<<1, 256, 0, stream>>>(X3, w3a, b3a, w3b, b3b, (float*)d_out);
}